// TuringMachine_22634477650699
// MI455X (gfx1250) — compile-verified
//
#include <hip/hip_runtime.h>
#include <hip/hip_bf16.h>

// ---------------------------------------------------------------------------
// Soft binary tree ("TuringMachine") forward in linear probability space.
// All logsumexp contractions are folded into f32 WMMA GEMMs on CDNA5:
//
//   P1 = w1*z0, P2 = w0*z0 + w1*z1, P3 = w0*z1 are each linear in
//   [exp(x0)|exp(x1)], so the per-level compare recurrence needs only
//   3 LDS loads + 3 FMAs per element:
//       S1 += S2*P1 ; S3 += S2*P3 ; S2 *= P2
//   Final output: m + log( exp(attn-m) @ exp(v)^T )  (WMMA, K=256)
//
// All WMMA fragment elements are contiguous 8B pairs -> b64 loads.
// ---------------------------------------------------------------------------

#define TM_B      8192
#define TM_DEPTH  8
#define TM_NCOL   1020          // sum over levels of 4*2^i, i=0..7
#define TM_NPAD   1024

#define XC_STRIDE 34            // even (b64-aligned rows), 34*m%64 conflict-free
#define AT_STRIDE 264           // even, != 0 mod 64

typedef float v2f __attribute__((ext_vector_type(2)));
typedef float v8f __attribute__((ext_vector_type(8)));
typedef int   v4i __attribute__((ext_vector_type(4)));

#if defined(__AMDGCN__) && __has_builtin(__builtin_amdgcn_global_load_async_to_lds_b128)
#define TM_ASYNC_LDS 1
typedef __attribute__((address_space(1))) v4i gv4i;   // global (AS1) 16B chunk
typedef __attribute__((address_space(3))) v4i lv4i;   // LDS    (AS3) 16B chunk
#endif

struct WPtrs {
    const float* sw[8];
    const float* dw[8];
    const float* cw[8];
    const float* vw;
};

// workspace layout (floats)
//   EP1 [1024][32]  coeffs of [X0|X1] for P1 = w1*z0          @ 0
//   EP2 [1024][32]  coeffs of [X0|X1] for P2 = w0*z0 + w1*z1  @ 32768
//   EP3 [1024][16]  coeffs of X1      for P3 = w0*z1          @ 65536
//   Vt  [16][256]   exp(v0)|exp(v1), N-major (transposed)     @ 81920
#define WS_EP1 0
#define WS_EP2 32768
#define WS_EP3 65536
#define WS_V   81920

__device__ __forceinline__ float lax(float a, float b) {
    float m = fmaxf(a, b);
    return m + log1pf(expf(-fabsf(a - b)));
}

// ---------------------------------------------------------------------------
// Kernel 1: pre-exponentiate and pre-combine all weights (batch independent).
// ---------------------------------------------------------------------------
__global__ void __launch_bounds__(256)
tm_precompute(WPtrs wp, float* __restrict__ ws)
{
    int gid = blockIdx.x * 256 + threadIdx.x;

    if (gid < TM_NPAD) {                       // per global column cc
        int cc = gid;
        float* EP1 = ws + WS_EP1;
        float* EP2 = ws + WS_EP2;
        float* EP3 = ws + WS_EP3;
        if (cc >= TM_NCOL) {                   // padding columns -> zeros
            for (int k = 0; k < 32; ++k) { EP1[cc * 32 + k] = 0.0f; EP2[cc * 32 + k] = 0.0f; }
            for (int k = 0; k < 16; ++k) EP3[cc * 16 + k] = 0.0f;
            return;
        }
        int q   = cc >> 2;                              // coff_i = 4*(2^i-1)
        int lvl = 31 - __builtin_clz(q + 1);            // q+1 in [2^i, 2^{i+1})
        int n4  = 4 << lvl;                             // columns in level
        int cl  = cc - 4 * ((1 << lvl) - 1);            // local column

        // shuffle weights -> e[k] = exp(y[c,k])
        const float* sw = wp.sw[lvl];                   // (2, n4, 4)
        float w0b[4], w1b[4];
        #pragma unroll
        for (int bit = 0; bit < 4; ++bit) {
            float a = sw[cl * 4 + bit];
            float b = sw[n4 * 4 + cl * 4 + bit];
            float z = lax(a, b);
            w0b[bit] = a - z;
            w1b[bit] = b - z;
        }
        // direction weights d0,d1 = softmax(dw)
        float d0, d1;
        {
            const float* dw = wp.dw[lvl];               // (2, n4)
            float a = dw[cl], b = dw[n4 + cl];
            float z = lax(a, b);
            d0 = expf(a - z); d1 = expf(b - z);
        }
        // compare weights w0,w1 = softmax(cw)
        float w0, w1;
        {
            const float* cw = wp.cw[lvl];               // (2, 1, n, 4) == (2, n4)
            float a = cw[cl], b = cw[n4 + cl];
            float z = lax(a, b);
            w0 = expf(a - z); w1 = expf(b - z);
        }
        for (int k = 0; k < 16; ++k) {
            float y = 0.0f;
            #pragma unroll
            for (int bit = 0; bit < 4; ++bit)
                y += ((k >> (3 - bit)) & 1) ? w0b[bit] : w1b[bit];
            float e  = expf(y);
            float e0 = d0 * e, e1 = d1 * e;
            EP1[cc * 32 + k]      = w1 * e0;                     // X0 coeff of P1
            EP1[cc * 32 + 16 + k] = w1 * e1;                     // X1 coeff of P1
            EP2[cc * 32 + k]      = w0 * e0;                     // X0 coeff of P2
            EP2[cc * 32 + 16 + k] = fmaf(2.0f * w1, e0, w0 * e1);// X1 coeff of P2
            EP3[cc * 16 + k]      = 2.0f * w0 * e0;              // X1 coeff of P3
        }
        return;
    }

    int g2 = gid - TM_NPAD;
    if (g2 < 2048) {                           // value weights: (j, f) pairs
        int j = g2 >> 3, f = g2 & 7;
        const float* vw = wp.vw;               // (2, 1, 256, 8)
        float a = vw[j * 8 + f];
        float b = vw[2048 + j * 8 + f];
        float z = lax(a, b);
        float* Vt = ws + WS_V;                 // [16][256]: rows 0..7 = v0, 8..15 = v1
        Vt[f * 256 + j]       = expf(a - z);
        Vt[(8 + f) * 256 + j] = expf(b - z);
    }
}

// ---------------------------------------------------------------------------
// Kernel 2: fused forward. One block = 16 batch rows, 128 threads (4 wave32).
// Dynamic LDS (floats):
//   sXc   [16][34]     544    [exp(x0)|exp(x1)] tile, even-padded rows
//   sXraw [512]        512    raw async-staged x0|x1 tile
//   sP1   [16][1024] 16384
//   sP2   [16][1024] 16384
//   sP3   [16][1024] 16384
//   sAttn [16][264]   4224
//   sM    [16]          16
// total 54,448 floats = 217,792 B  (fits 320KB/WGP)
// ---------------------------------------------------------------------------
__global__ void __launch_bounds__(128)
tm_forward(const float* __restrict__ x0, const float* __restrict__ x1,
           const float* __restrict__ ws, float* __restrict__ out)
{
    extern __shared__ float smem[];
    float* sXc   = smem;                // 544
    float* sXraw = smem + 544;          // 512
    float* sP1   = smem + 1056;
    float* sP2   = sP1 + 16384;
    float* sP3   = sP2 + 16384;
    float* sAttn = sP3 + 16384;
    float* sM    = sAttn + 16 * AT_STRIDE;

    const float* EP1g = ws + WS_EP1;
    const float* EP2g = ws + WS_EP2;
    const float* EP3g = ws + WS_EP3;
    const float* Vtg  = ws + WS_V;

    const int tid   = threadIdx.x;
    const int lane  = tid & 31;
    const int wave  = tid >> 5;
    const int half  = (lane >> 4) & 1;       // WMMA lane-half (K pair select)
    const int ln    = lane & 15;
    const int h2    = 2 * half;              // fragment pair offset
    const int bbase = blockIdx.x * 16;

    __builtin_prefetch(EP1g, 0, 3);          // global_prefetch_b8 (L2-hot weights)
    __builtin_prefetch(EP2g, 0, 3);

    // ---- stage raw x tiles (contiguous 1KB each) into LDS ------------------
#if defined(TM_ASYNC_LDS)
    {   // 128 b128 chunks: one per thread, tracked by ASYNCcnt
        int mat   = tid >> 6;                // 0 -> x0, 1 -> x1
        int chunk = tid & 63;                // 64 x 16B per matrix
        const float* gsrc = (mat ? x1 : x0) + bbase * 16 + chunk * 4;
        float*       ldst = sXraw + mat * 256 + chunk * 4;
        __builtin_amdgcn_global_load_async_to_lds_b128((gv4i*)gsrc, (lv4i*)ldst, 0, 0);
#if __has_builtin(__builtin_amdgcn_s_wait_asynccnt)
        __builtin_amdgcn_s_wait_asynccnt(0);
#else
        asm volatile("s_wait_asynccnt 0" ::: "memory");
#endif
    }
    __syncthreads();
    for (int idx = tid; idx < 512; idx += 128) {
        int mat = idx >> 8, r = idx & 255;
        int m = r >> 4, k = r & 15;
        sXc[m * XC_STRIDE + mat * 16 + k] = expf(sXraw[idx]);
    }
#else
    for (int idx = tid; idx < 512; idx += 128) {
        int mat = idx >> 8, r = idx & 255;
        int m = r >> 4, k = r & 15;
        sXc[m * XC_STRIDE + mat * 16 + k] = expf((mat ? x1 : x0)[(bbase + m) * 16 + k]);
    }
#endif
    if (tid < 16) sAttn[tid * AT_STRIDE] = 0.0f;  // attn starts as (B,1) zeros
    __syncthreads();

    // ---- phase 1: P1,P2 = [X0|X1] @ EP{1,2}^T (K=32), P3 = X1 @ EP3^T ------
    // 64 N-tiles over 4 waves, 20 chained v_wmma_f32_16x16x4_f32 per tile.
    // Fragment pairs are contiguous 8B -> single b64 loads. EXEC all-ones.
    for (int nt = wave; nt < 64; nt += 4) {
        int nb = nt * 16;
        v8f a1 = {0.f,0.f,0.f,0.f,0.f,0.f,0.f,0.f};
        v8f a2 = a1, a3 = a1;
        #pragma unroll
        for (int kc = 0; kc < 8; ++kc) {     // shared A-fragment for P1/P2
            int kb = kc * 4;
            v2f a  = *(const v2f*)(sXc  + ln * XC_STRIDE + kb + h2);
            v2f b1 = *(const v2f*)(EP1g + (nb + ln) * 32 + kb + h2);
            v2f b2 = *(const v2f*)(EP2g + (nb + ln) * 32 + kb + h2);
            a1 = __builtin_amdgcn_wmma_f32_16x16x4_f32(false, a, false, b1, (short)0, a1, false, false);
            a2 = __builtin_amdgcn_wmma_f32_16x16x4_f32(false, a, false, b2, (short)0, a2, false, false);
        }
        #pragma unroll
        for (int kc = 0; kc < 4; ++kc) {     // P3: X1 half only (K=16)
            int kb = kc * 4;
            v2f a = *(const v2f*)(sXc  + ln * XC_STRIDE + 16 + kb + h2);
            v2f b = *(const v2f*)(EP3g + (nb + ln) * 16 + kb + h2);
            a3 = __builtin_amdgcn_wmma_f32_16x16x4_f32(false, a, false, b, (short)0, a3, false, false);
        }
        #pragma unroll
        for (int r = 0; r < 8; ++r) {        // D: VGPR r -> rows r / r+8
            int row = (r + 8 * half) * 1024 + nb + ln;
            sP1[row] = a1[r];
            sP2[row] = a2[r];
            sP3[row] = a3[r];
        }
    }
    __syncthreads();

    // ---- phase 2: tree levels, 3 LDS loads + 3 FMAs per element ------------
    const float W1C = 1.0f;                  // exp(BOUND - logaddexp(B,-B)) ~= 1
    const float W2C = 1.8622957e-44f;        // exp(-2*BOUND - log2): ~0 (denorm)
    for (int lvl = 0; lvl < TM_DEPTH; ++lvl) {
        int n    = 1 << lvl;
        int coff = 4 * (n - 1);
        for (int idx = tid; idx < 16 * n; idx += 128) {
            int b  = idx >> lvl;
            int j  = idx & (n - 1);
            int cb = b * 1024 + coff + j * 4;
            float S1 = W2C, S2 = W1C, S3 = W2C;
            #pragma unroll
            for (int s = 0; s < 4; ++s) {
                float p1 = sP1[cb + s], p2 = sP2[cb + s], p3 = sP3[cb + s];
                S1 = fmaf(S2, p1, S1);
                S3 = fmaf(S2, p3, S3);
                S2 = S2 * p2;
            }
            float base = sAttn[b * AT_STRIDE + j];
            sAttn[b * AT_STRIDE + j]     = base + logf(S1 + S2);
            sAttn[b * AT_STRIDE + j + n] = base + logf(S3);
        }
        __syncthreads();
    }

    // ---- phase 3: per-row max + exp(attn - max) ----------------------------
    if (tid < 16) {
        float m = -3.4e38f;
        for (int j = 0; j < 256; ++j) m = fmaxf(m, sAttn[tid * AT_STRIDE + j]);
        sM[tid] = m;
    }
    __syncthreads();
    for (int idx = tid; idx < 4096; idx += 128) {
        int b = idx >> 8, j = idx & 255;
        sAttn[b * AT_STRIDE + j] = expf(sAttn[b * AT_STRIDE + j] - sM[b]);
    }
    __syncthreads();

    // ---- phase 4: out = m + log( exp(attn-m) @ V^T ), WMMA over K=256 ------
    if (wave == 0) {                          // whole wave -> EXEC all-ones
        v8f acc = {0.f,0.f,0.f,0.f,0.f,0.f,0.f,0.f};
        for (int kc = 0; kc < 64; ++kc) {
            int kb = kc * 4;
            v2f a = *(const v2f*)(sAttn + ln * AT_STRIDE + kb + h2);
            v2f b = *(const v2f*)(Vtg   + ln * 256       + kb + h2);
            acc = __builtin_amdgcn_wmma_f32_16x16x4_f32(false, a, false, b, (short)0, acc, false, false);
        }
        int t = ln >> 3;                      // 0 -> y0 block, 1 -> y1 block
        int f = ln & 7;
        #pragma unroll
        for (int r = 0; r < 8; ++r) {
            int m = r + 8 * half;
            out[t * TM_B * 8 + (bbase + m) * 8 + f] = logf(acc[r]) + sM[m];
        }
    }
}

// ---------------------------------------------------------------------------
extern "C" void kernel_launch(void* const* d_in, const int* in_sizes, int n_in,
                              void* d_out, int out_size, void* d_ws, size_t ws_size,
                              hipStream_t stream)
{
    (void)in_sizes; (void)n_in; (void)out_size; (void)ws_size;
    const float* x0 = (const float*)d_in[0];
    const float* x1 = (const float*)d_in[1];

    WPtrs wp;
    for (int i = 0; i < 8; ++i) {            // dict order: sw_i, dw_i, cw_i
        wp.sw[i] = (const float*)d_in[2 + 3 * i];
        wp.dw[i] = (const float*)d_in[3 + 3 * i];
        wp.cw[i] = (const float*)d_in[4 + 3 * i];
    }
    wp.vw = (const float*)d_in[26];

    float* ws = (float*)d_ws;                // 344,064 bytes used

    tm_precompute<<<12, 256, 0, stream>>>(wp, ws);

    const size_t smem = 54448 * sizeof(float);   // 217,792 B dynamic LDS
    tm_forward<<<TM_B / 16, 128, smem, stream>>>(x0, x1, ws, (float*)d_out);
}